// GradPeak_20229295964375
// MI455X (gfx1250) — compile-verified
//
#include <hip/hip_runtime.h>

#define B_      32
#define N_      131072
#define PLEN    (N_ - 1)
#define ECHO    32
#define MAXC    65536
#define THRESH  0.01f
#define MTHRESH (-0.0025f)
#define RADIUS_ 10
#define SIGMA_  (19.0f / 6.0f)
#define IVMIN   12
#define IVMAX   600
#define PI_F    3.14159265358979323846f

typedef float v2f __attribute__((ext_vector_type(2)));
typedef float v8f __attribute__((ext_vector_type(8)));

// ---------------------------------------------------------------------------
// Radix-16 Stockham FFT pass. One wave handles 16 butterflies as a 16x16
// complex tile: D = DFT16 * V, computed with v_wmma_f32_16x16x4_f32
// (4 K-chunks x 4 real matmuls = 16 WMMA per tile).
// mode: 0 = complex in, 1 = real input (im = 0), 2 = hilbert-mask + conjugate.
// ---------------------------------------------------------------------------
__global__ __launch_bounds__(256)
void fft16_pass(const float* __restrict__ inr, const float* __restrict__ ini,
                float* __restrict__ outr, float* __restrict__ outi,
                int Ns, int mode)
{
    const int lane = threadIdx.x & 31;
    const int wave = threadIdx.x >> 5;
    const int half = lane >> 4;        // 0: lanes 0-15, 1: lanes 16-31
    const int col  = lane & 15;        // butterfly column within tile / DFT row m
    const int tile = blockIdx.x * 8 + wave;
    const int row  = blockIdx.y;
    const int j    = tile * 16 + col;  // butterfly index
    const long base   = (long)row * N_;
    const int  stride = N_ / 16;       // 8192

    v8f Dr = {};
    v8f Di = {};

    const int   jmod   = j % Ns;
    const float twbase = -2.0f * PI_F / (16.0f * (float)Ns);
    const int   twmod  = 16 * Ns;

    #pragma unroll
    for (int c = 0; c < 4; ++c) {
        v2f Ar, AiP, AiN, Br, Bi;
        #pragma unroll
        for (int v = 0; v < 2; ++v) {
            const int k = 4 * c + v + 2 * half;   // K index (input point)
            // DFT16[m][k] = exp(-2*pi*i*m*k/16), m = col (integer range reduction)
            const int   r16  = (col * k) & 15;
            const float aang = -(PI_F / 8.0f) * (float)r16;
            const float ac = __cosf(aang), as = __sinf(aang);
            Ar[v]  = ac;
            AiP[v] = as;
            AiN[v] = -as;
            // load V[k][col]
            const long idx = base + (long)j + (long)k * stride;
            float vr, vi;
            if (mode == 1) {
                vr = inr[idx]; vi = 0.0f;
            } else if (mode == 2) {
                const int n = j + k * stride;     // natural-order spectral index
                const float h = (n == 0 || n == N_ / 2) ? 1.0f
                              : (n < N_ / 2 ? 2.0f : 0.0f);
                vr =  h * inr[idx];
                vi = -h * ini[idx];               // conjugate for inverse-via-forward
            } else {
                vr = inr[idx]; vi = ini[idx];
            }
            // twiddle W^(k*(j%Ns)), W = exp(-2*pi*i/(16*Ns))
            const int   kj   = (k * jmod) % twmod;
            const float tang = twbase * (float)kj;
            const float tc = __cosf(tang), ts = __sinf(tang);
            Br[v] = vr * tc - vi * ts;
            Bi[v] = vr * ts + vi * tc;
        }
        // Dr += Ar*Br - Ai*Bi ; Di += Ar*Bi + Ai*Br  (A-negation folded into AiN)
        Dr = __builtin_amdgcn_wmma_f32_16x16x4_f32(false, Ar,  false, Br, (short)0, Dr, false, false);
        Dr = __builtin_amdgcn_wmma_f32_16x16x4_f32(false, AiN, false, Bi, (short)0, Dr, false, false);
        Di = __builtin_amdgcn_wmma_f32_16x16x4_f32(false, Ar,  false, Bi, (short)0, Di, false, false);
        Di = __builtin_amdgcn_wmma_f32_16x16x4_f32(false, AiP, false, Br, (short)0, Di, false, false);
    }

    // Stockham scatter: out[(j/Ns)*16*Ns + j%Ns + i*Ns]; lane holds rows i = g + 8*half
    const long dbase = base + (long)(j / Ns) * (Ns * 16) + jmod;
    #pragma unroll
    for (int g = 0; g < 8; ++g) {
        const int i = g + 8 * half;
        outr[dbase + (long)i * Ns] = Dr[g];
        outi[dbase + (long)i * Ns] = Di[g];
    }
}

// Final radix-2 pass (Ns = 65536).
__global__ __launch_bounds__(256)
void fft2_pass(const float* __restrict__ inr, const float* __restrict__ ini,
               float* __restrict__ outr, float* __restrict__ outi, int Ns)
{
    const int  j    = blockIdx.x * blockDim.x + threadIdx.x;   // 0 .. N/2-1
    const long base = (long)blockIdx.y * N_;
    const int  halfN = N_ / 2;
    const float v0r = inr[base + j],         v0i = ini[base + j];
    const float v1r = inr[base + j + halfN], v1i = ini[base + j + halfN];
    const int   jm  = j % Ns;
    const float ang = -PI_F * (float)jm / (float)Ns;
    const float c = __cosf(ang), s = __sinf(ang);
    const float t1r = v1r * c - v1i * s;
    const float t1i = v1r * s + v1i * c;
    const long d = base + (long)(j / Ns) * (2 * Ns) + jm;
    outr[d]      = v0r + t1r;  outi[d]      = v0i + t1i;
    outr[d + Ns] = v0r - t1r;  outi[d + Ns] = v0i - t1i;
}

// env = |ifft(H*X)| = |fft(conj(H*X))| / N
__global__ __launch_bounds__(256)
void envelope_k(const float* __restrict__ re, const float* __restrict__ im,
                float* __restrict__ env)
{
    const long i = (long)blockIdx.y * N_ + blockIdx.x * blockDim.x + threadIdx.x;
    const float r = re[i], m = im[i];
    env[i] = sqrtf(r * r + m * m) * (1.0f / (float)N_);
}

__device__ __forceinline__ float grad_at(const float* __restrict__ env, long base, int m)
{
    if (m < 0 || m >= N_) return 0.0f;                     // conv zero padding
    if (m == 0)       return (env[base + 1] - env[base]) * 0.1f;
    if (m == N_ - 1)  return (env[base + N_ - 1] - env[base + N_ - 2]) * 0.1f;
    return (env[base + m + 1] - env[base + m - 1]) * 0.05f;
}

// torch.gradient + gaussian_filter_1d fused
__global__ __launch_bounds__(256)
void gradsmooth_k(const float* __restrict__ env, float* __restrict__ gs)
{
    __shared__ float kg[2 * RADIUS_ + 1];
    __shared__ float kinv;
    const int t = threadIdx.x;
    if (t < 2 * RADIUS_ + 1) {
        const float d = (float)(t - RADIUS_) / SIGMA_;
        kg[t] = __expf(-0.5f * d * d);
    }
    __syncthreads();
    if (t == 0) {
        float s = 0.0f;
        for (int i = 0; i < 2 * RADIUS_ + 1; ++i) s += kg[i];
        kinv = 1.0f / s;
    }
    __syncthreads();
    const int  n    = blockIdx.x * blockDim.x + t;
    const long base = (long)blockIdx.y * N_;
    float acc = 0.0f;
    #pragma unroll
    for (int d = 0; d <= 2 * RADIUS_; ++d)
        acc += kg[d] * grad_at(env, base, n + d - RADIUS_);
    gs[base + n] = acc * kinv;
}

__global__ void zero_cnt(int* c) { if (threadIdx.x < B_) c[threadIdx.x] = 0; }

// ---------------------------------------------------------------------------
// Edge matching + dedup. last_plus is monotone non-decreasing in m, so
// "first minus edge per unique plus edge" is a segmented-min scan keyed by
// last_plus. Two LDS Hillis-Steele scans per 1024-chunk with carries.
// ---------------------------------------------------------------------------
__global__ __launch_bounds__(1024)
void detect_k(const float* __restrict__ gs, const float* __restrict__ env,
              int* __restrict__ cand_m, float* __restrict__ cand_amp,
              int* __restrict__ cand_cnt)
{
    __shared__ int s[1024];     // last-plus max-scan
    __shared__ int skey[1024];  // segment key (final lp)
    __shared__ int smv[1024];   // segmented min of valid m
    const int  t    = threadIdx.x;
    const int  row  = blockIdx.x;
    const long base = (long)row * N_;

    int carry_lp = -1;
    int ckey = -2;              // carry pair for segmented min (never matches)
    int cmv  = 0x7fffffff;

    const int CHUNKS = (PLEN + 1023) / 1024;   // 128
    for (int ch = 0; ch < CHUNKS; ++ch) {
        const int  m  = ch * 1024 + t;
        const bool ir = (m < PLEN);
        const float g0 = ir ? gs[base + m]     : 0.0f;
        const float g1 = ir ? gs[base + m + 1] : 0.0f;
        const bool pe = ir && (g1 > THRESH)  && !(g0 > THRESH);
        const bool me = ir && (g1 < MTHRESH) && !(g0 < MTHRESH);

        // inclusive max-scan for last plus edge
        s[t] = pe ? m : -1;
        __syncthreads();
        for (int off = 1; off < 1024; off <<= 1) {
            const int v = s[t];
            const int a = (t >= off) ? s[t - off] : -1;
            __syncthreads();
            s[t] = (a > v) ? a : v;
            __syncthreads();
        }
        const int lp  = (carry_lp > s[t]) ? carry_lp : s[t];
        const int gap = m - lp;
        const bool valid = me && lp >= 0 && gap > IVMIN && gap < IVMAX;

        // segmented min-scan: first valid m within each lp-segment
        skey[t] = lp;
        smv[t]  = valid ? m : 0x7fffffff;
        __syncthreads();
        for (int off = 1; off < 1024; off <<= 1) {
            const int bk = skey[t];
            const int bm = smv[t];
            const int ak = (t >= off) ? skey[t - off] : -3;
            const int am = (t >= off) ? smv[t - off]  : 0x7fffffff;
            __syncthreads();
            smv[t] = (ak == bk && am < bm) ? am : bm;
            __syncthreads();
        }
        const int fk = skey[t];
        const int fm = smv[t];
        const int finalmv = (ckey == fk && cmv < fm) ? cmv : fm;
        const bool kept = valid && (finalmv == m);
        if (kept) {
            const int pos = atomicAdd(&cand_cnt[row], 1);
            if (pos < MAXC) {
                cand_m[row * MAXC + pos]   = m;
                cand_amp[row * MAXC + pos] = env[base + m];   // amplitude at minus edge
            }
        }
        __syncthreads();
        // update carries from chunk tails
        const int lastS = s[1023];
        const int lastK = skey[1023];
        const int lastM = smv[1023];
        const int nm = (ckey == lastK && cmv < lastM) ? cmv : lastM;
        ckey = lastK; cmv = nm;
        carry_lp = (lastS > carry_lp) ? lastS : carry_lp;
        __syncthreads();
    }
}

// top ECHO by amplitude (tie -> smaller index, matching lax.top_k stability),
// zero-fill missing, ascending sort of indices.
__global__ __launch_bounds__(256)
void topk_k(const int* __restrict__ cand_m, float* __restrict__ cand_amp,
            const int* __restrict__ cand_cnt, float* __restrict__ out)
{
    __shared__ float ra[256];
    __shared__ int   rm[256];
    __shared__ int   rs[256];
    __shared__ int   sel[ECHO];
    const int t   = threadIdx.x;
    const int row = blockIdx.x;
    int cnt = cand_cnt[row];
    if (cnt > MAXC) cnt = MAXC;
    const int* cm = cand_m  + row * MAXC;
    float*     ca = cand_amp + row * MAXC;

    for (int it = 0; it < ECHO; ++it) {
        float ba = -2.0e30f;
        int   bm = 0x7fffffff;
        int   bs = -1;
        for (int i = t; i < cnt; i += 256) {
            const float a = ca[i];
            if (a < -9.0e29f) continue;                    // invalidated slot
            const int mm = cm[i];
            if (a > ba || (a == ba && mm < bm)) { ba = a; bm = mm; bs = i; }
        }
        ra[t] = ba; rm[t] = bm; rs[t] = bs;
        __syncthreads();
        for (int off = 128; off > 0; off >>= 1) {
            if (t < off) {
                const float a2 = ra[t + off];
                const int   m2 = rm[t + off];
                if (a2 > ra[t] || (a2 == ra[t] && m2 < rm[t])) {
                    ra[t] = a2; rm[t] = m2; rs[t] = rs[t + off];
                }
            }
            __syncthreads();
        }
        if (t == 0) {
            if (rs[0] >= 0) { sel[it] = rm[0]; ca[rs[0]] = -1.0e30f; }
            else            { sel[it] = 0; }               // pad slot
        }
        __syncthreads();
    }
    if (t == 0) {
        for (int i = 1; i < ECHO; ++i) {                   // ascending insertion sort
            const int v = sel[i];
            int j = i - 1;
            while (j >= 0 && sel[j] > v) { sel[j + 1] = sel[j]; --j; }
            sel[j + 1] = v;
        }
        for (int i = 0; i < ECHO; ++i) out[row * ECHO + i] = (float)sel[i];
    }
}

extern "C" void kernel_launch(void* const* d_in, const int* in_sizes, int n_in,
                              void* d_out, int out_size, void* d_ws, size_t ws_size,
                              hipStream_t stream)
{
    const float* x = (const float*)d_in[0];
    float* out = (float*)d_out;

    const size_t BN = (size_t)B_ * N_;
    float* b0r = (float*)d_ws;
    float* b0i = b0r + BN;
    float* b1r = b0i + BN;
    float* b1i = b1r + BN;
    int*   cnt = (int*)(b1i + BN);
    // buffer reuse after the FFTs finish:
    float* env      = b0r;
    float* gs       = b0i;
    int*   cand_m   = (int*)b1r;      // 32 * 65536 ints  <= BN
    float* cand_amp = b1i;

    const dim3 blk(256);
    const dim3 g16(64, B_);           // 512 tiles/row, 8 waves/block
    const dim3 g2(256, B_);           // 65536 butterflies/row
    const dim3 gE(512, B_);

    // forward FFT of x (real load fused into pass 1)
    fft16_pass<<<g16, blk, 0, stream>>>(x,   nullptr, b0r, b0i,    1, 1);
    fft16_pass<<<g16, blk, 0, stream>>>(b0r, b0i,     b1r, b1i,   16, 0);
    fft16_pass<<<g16, blk, 0, stream>>>(b1r, b1i,     b0r, b0i,  256, 0);
    fft16_pass<<<g16, blk, 0, stream>>>(b0r, b0i,     b1r, b1i, 4096, 0);
    fft2_pass <<<g2,  blk, 0, stream>>>(b1r, b1i,     b0r, b0i, 65536);
    // inverse FFT of H*X via forward FFT of conj(H*X) (mask+conj fused into pass 1)
    fft16_pass<<<g16, blk, 0, stream>>>(b0r, b0i,     b1r, b1i,    1, 2);
    fft16_pass<<<g16, blk, 0, stream>>>(b1r, b1i,     b0r, b0i,   16, 0);
    fft16_pass<<<g16, blk, 0, stream>>>(b0r, b0i,     b1r, b1i,  256, 0);
    fft16_pass<<<g16, blk, 0, stream>>>(b1r, b1i,     b0r, b0i, 4096, 0);
    fft2_pass <<<g2,  blk, 0, stream>>>(b0r, b0i,     b1r, b1i, 65536);

    envelope_k  <<<gE, blk, 0, stream>>>(b1r, b1i, env);
    gradsmooth_k<<<gE, blk, 0, stream>>>(env, gs);
    zero_cnt    <<<1, 32, 0, stream>>>(cnt);
    detect_k    <<<B_, 1024, 0, stream>>>(gs, env, cand_m, cand_amp, cnt);
    topk_k      <<<B_, 256, 0, stream>>>(cand_m, cand_amp, cnt, out);
}